// TwoLayerGCN_31095563223114
// MI455X (gfx1250) — compile-verified
//
#include <hip/hip_runtime.h>
#include <hip/hip_bf16.h>

typedef float v2f __attribute__((ext_vector_type(2)));
typedef float v4f __attribute__((ext_vector_type(4)));
typedef float v8f __attribute__((ext_vector_type(8)));

#define DIM 128
#define ODIM 64

// ---------------- degree / inverse degree ----------------
__global__ __launch_bounds__(256) void deg_count(const int* __restrict__ dst,
                                                 float* __restrict__ deg, int n_edges) {
    int e = blockIdx.x * blockDim.x + threadIdx.x;
    if (e < n_edges) atomicAdd(&deg[dst[e]], 1.0f);
}

__global__ __launch_bounds__(256) void inv_deg_k(const float* __restrict__ deg,
                                                 float* __restrict__ invd, int n) {
    int i = blockIdx.x * blockDim.x + threadIdx.x;
    if (i < n) invd[i] = 1.0f / fmaxf(deg[i], 1.0f);
}

// ---------------- weight transpose: W[DIM][nout] -> WT[nout][DIM] ------------
__global__ __launch_bounds__(256) void transpose_w(const float* __restrict__ Ws,
                                                   const float* __restrict__ Wn,
                                                   float* __restrict__ WsT,
                                                   float* __restrict__ WnT,
                                                   int nout) {
    int i = blockIdx.x * blockDim.x + threadIdx.x;
    if (i < DIM * nout) {
        int k = i / nout, c = i - k * nout;
        WsT[c * DIM + k] = Ws[i];
        WnT[c * DIM + k] = Wn[i];
    }
}

// ---------------- edge scatter-add (mean numerator) ----------------
__global__ __launch_bounds__(256) void scatter_add(const float* __restrict__ h,
                                                   const int* __restrict__ src,
                                                   const int* __restrict__ dst,
                                                   float* __restrict__ m, int n_edges) {
    int wave = blockIdx.x * (blockDim.x >> 5) + (threadIdx.x >> 5);
    int lane = threadIdx.x & 31;
    if (wave >= n_edges) return;
    int s = src[wave];
    int d = dst[wave];
    const float4 v = *(const float4*)(h + (size_t)s * DIM + lane * 4);
    float* mp = m + (size_t)d * DIM + lane * 4;
    atomicAdd(mp + 0, v.x);
    atomicAdd(mp + 1, v.y);
    atomicAdd(mp + 2, v.z);
    atomicAdd(mp + 3, v.w);
}

// ---------------- fused SAGE layer GEMM via V_WMMA_F32_16X16X4_F32 ----------
// One wave computes a 64x16 tile of out = relu(h@Ws + diag(invd)*(m@Wn) + b).
// K processed in permuted 8-wide chunks: each lane's data for two WMMA steps
// is ONE contiguous b128 load (same K-permutation applied to A and B).
// Fully-unrolled software pipeline: chunk j+1's 10 b128 loads (all at
// immediate offsets from 10 base pointers) issue before chunk j's 16 WMMAs.
// inv_deg applied in the epilogue (row scaling commutes with right-mult).
template <int NOUT>
__global__ __launch_bounds__(32) void sage_gemm(const float* __restrict__ h,
                                                const float* __restrict__ m,
                                                const float* __restrict__ invd,
                                                const float* __restrict__ WsT,
                                                const float* __restrict__ WnT,
                                                const float* __restrict__ bias,
                                                float* __restrict__ out,
                                                int n_rows) {
    const int lane = threadIdx.x;            // full wave active: WMMA needs EXEC all-1s
    const int r    = lane & 15;
    const int koff = (lane >> 4) << 2;       // 0 or 4 floats (upper-half lanes)
    const int row0 = blockIdx.x << 6;        // 64 rows per wave
    const int c    = (blockIdx.y << 4) + r;  // output column for this lane

    const float* hrow[4];
    const float* mrow[4];
#pragma unroll
    for (int t = 0; t < 4; ++t) {
        int rr = row0 + t * 16 + r;
        if (rr >= n_rows) rr = n_rows - 1;   // clamp: EXEC stays uniform
        hrow[t] = h + (size_t)rr * DIM + koff;
        mrow[t] = m + (size_t)rr * DIM + koff;
    }
    const float* wsb = WsT + (size_t)c * DIM + koff;
    const float* wnb = WnT + (size_t)c * DIM + koff;

    v8f acc_s[4] = {};
    v8f acc_n[4] = {};

    // ---- prologue: load chunk 0 (immediate offset 0) ----
    v4f cBs = *(const v4f*)(wsb);
    v4f cBn = *(const v4f*)(wnb);
    v4f cAh[4], cAm[4];
#pragma unroll
    for (int t = 0; t < 4; ++t) {
        cAh[t] = *(const v4f*)(hrow[t]);
        cAm[t] = *(const v4f*)(mrow[t]);
    }

#pragma unroll
    for (int j = 0; j < 16; ++j) {
        v4f nBs, nBn, nAh[4], nAm[4];
        if (j < 15) {
            const int kn = (j + 1) * 8;      // compile-time: folds into IOFFSET
            nBs = *(const v4f*)(wsb + kn);
            nBn = *(const v4f*)(wnb + kn);
#pragma unroll
            for (int t = 0; t < 4; ++t) {
                nAh[t] = *(const v4f*)(hrow[t] + kn);
                nAm[t] = *(const v4f*)(mrow[t] + kn);
            }
        }
        // ---- 16 WMMAs on current chunk (8 independent acc chains) ----
        v2f bs0 = {cBs.x, cBs.y}, bs1 = {cBs.z, cBs.w};
        v2f bn0 = {cBn.x, cBn.y}, bn1 = {cBn.z, cBn.w};
#pragma unroll
        for (int t = 0; t < 4; ++t) {
            v2f ah0 = {cAh[t].x, cAh[t].y}, ah1 = {cAh[t].z, cAh[t].w};
            v2f am0 = {cAm[t].x, cAm[t].y}, am1 = {cAm[t].z, cAm[t].w};
            acc_s[t] = __builtin_amdgcn_wmma_f32_16x16x4_f32(false, ah0, false, bs0,
                                                             (short)0, acc_s[t], false, false);
            acc_n[t] = __builtin_amdgcn_wmma_f32_16x16x4_f32(false, am0, false, bn0,
                                                             (short)0, acc_n[t], false, false);
            acc_s[t] = __builtin_amdgcn_wmma_f32_16x16x4_f32(false, ah1, false, bs1,
                                                             (short)0, acc_s[t], false, false);
            acc_n[t] = __builtin_amdgcn_wmma_f32_16x16x4_f32(false, am1, false, bn1,
                                                             (short)0, acc_n[t], false, false);
        }
        if (j < 15) {
            cBs = nBs; cBn = nBn;            // SSA-renamed away by full unroll
#pragma unroll
            for (int t = 0; t < 4; ++t) { cAh[t] = nAh[t]; cAm[t] = nAm[t]; }
        }
    }

    // ---- epilogue: bias + inv_deg scaling + relu ----
    const float bv = bias[c];
    if (row0 + 64 <= n_rows) {               // fast path: no per-element predication
#pragma unroll
        for (int t = 0; t < 4; ++t) {
            const int rbase = row0 + t * 16 + ((lane >> 4) << 3);
#pragma unroll
            for (int v = 0; v < 8; ++v) {
                const int rr = rbase + v;
                float o = acc_s[t][v] + invd[rr] * acc_n[t][v] + bv;
                out[(size_t)rr * NOUT + c] = o > 0.0f ? o : 0.0f;
            }
        }
    } else {
#pragma unroll
        for (int t = 0; t < 4; ++t) {
            const int rbase = row0 + t * 16 + ((lane >> 4) << 3);
#pragma unroll
            for (int v = 0; v < 8; ++v) {
                const int rr = rbase + v;
                if (rr < n_rows) {
                    float o = acc_s[t][v] + invd[rr] * acc_n[t][v] + bv;
                    out[(size_t)rr * NOUT + c] = o > 0.0f ? o : 0.0f;
                }
            }
        }
    }
}

// ---------------- host-side orchestration ----------------
extern "C" void kernel_launch(void* const* d_in, const int* in_sizes, int n_in,
                              void* d_out, int out_size, void* d_ws, size_t ws_size,
                              hipStream_t stream) {
    const float* x          = (const float*)d_in[0];
    const int*   src        = (const int*)d_in[1];
    const int*   dst        = (const int*)d_in[2];
    const float* Wself      = (const float*)d_in[3];   // [7,128,128]
    const float* Wneigh     = (const float*)d_in[4];   // [7,128,128]
    const float* b          = (const float*)d_in[5];   // [7,128]
    const float* Wself_out  = (const float*)d_in[6];   // [128,64]
    const float* Wneigh_out = (const float*)d_in[7];   // [128,64]
    const float* b_out      = (const float*)d_in[8];   // [64]

    const int n = in_sizes[0] / DIM;   // 100000
    const int e = in_sizes[1];         // 1600000

    const size_t NH = (size_t)n * DIM;
    float* h_a  = (float*)d_ws;        // ping
    float* h_b  = h_a + NH;            // pong
    float* mbuf = h_b + NH;            // edge-sum accumulator (also deg scratch)
    float* invd = mbuf + NH;           // [n]
    float* wsT  = invd + n;            // transposed self weights
    float* wnT  = wsT + DIM * DIM;     // transposed neighbor weights
    (void)ws_size; (void)n_in; (void)out_size;

    // degrees -> inv_deg (reuse mbuf head as degree scratch)
    hipMemsetAsync(mbuf, 0, (size_t)n * sizeof(float), stream);
    deg_count<<<(e + 255) / 256, 256, 0, stream>>>(dst, mbuf, e);
    inv_deg_k<<<(n + 255) / 256, 256, 0, stream>>>(mbuf, invd, n);

    const int row_tiles   = (n + 63) / 64;   // 1563 (last block row-clamped)
    const int edge_blocks = (e + 7) / 8;     // 8 waves of 32 per block

    const float* hin = x;
    float* bufs[2] = {h_a, h_b};
    for (int layer = 0; layer < 7; ++layer) {
        float* hout = bufs[layer & 1];
        hipMemsetAsync(mbuf, 0, NH * sizeof(float), stream);
        transpose_w<<<(DIM * DIM + 255) / 256, 256, 0, stream>>>(
            Wself  + (size_t)layer * DIM * DIM,
            Wneigh + (size_t)layer * DIM * DIM, wsT, wnT, DIM);
        scatter_add<<<edge_blocks, 256, 0, stream>>>(hin, src, dst, mbuf, e);
        sage_gemm<DIM><<<dim3(row_tiles, DIM / 16), 32, 0, stream>>>(
            hin, mbuf, invd, wsT, wnT, b + (size_t)layer * DIM, hout, n);
        hin = hout;
    }

    // output layer: 128 -> 64
    hipMemsetAsync(mbuf, 0, NH * sizeof(float), stream);
    transpose_w<<<(DIM * ODIM + 255) / 256, 256, 0, stream>>>(
        Wself_out, Wneigh_out, wsT, wnT, ODIM);
    scatter_add<<<edge_blocks, 256, 0, stream>>>(hin, src, dst, mbuf, e);
    sage_gemm<ODIM><<<dim3(row_tiles, ODIM / 16), 32, 0, stream>>>(
        hin, mbuf, invd, wsT, wnT, b_out, (float*)d_out, n);
}